// NeighborhoodConsensus_89575837925966
// MI455X (gfx1250) — compile-verified
//
#include <hip/hip_runtime.h>

typedef __attribute__((ext_vector_type(2))) float v2f;
typedef __attribute__((ext_vector_type(8))) float v8f;

#define S    24
#define S2   576
#define S3   13824
#define S4   331776      // 24^4
#define SP   26
#define SP2  676
#define SP3  17576
#define SP4  456976      // 26^4  (zero-padded halo grid)

__device__ __forceinline__ int perm_tap(int t, int perm) {
    // t = ((dw*3+dx)*3+dy)*3+dz ; permuted branch reads w[.., dy,dz,dw,dx]
    if (!perm) return t;
    int dz = t % 3, dy = (t / 3) % 3, dx = (t / 9) % 3, dw = t / 27;
    return ((dy * 3 + dz) * 3 + dw) * 3 + dx;
}

// ---------------------------------------------------------------------------
// Zero both padded activation buffers (32*SP4 floats starting at actA).
// ---------------------------------------------------------------------------
__global__ __launch_bounds__(256)
void zero_ws(float4* __restrict__ p, int n4)
{
    for (int i = blockIdx.x * 256 + threadIdx.x; i < n4; i += gridDim.x * 256)
        p[i] = make_float4(0.f, 0.f, 0.f, 0.f);
}

// ---------------------------------------------------------------------------
// Copy x (8,24^4) into zero-padded xpad (8,26^4).
// ---------------------------------------------------------------------------
__global__ __launch_bounds__(256)
void pad_x(const float* __restrict__ x, float* __restrict__ xp)
{
    const int N = 8 * SP4;
    for (int i = blockIdx.x * 256 + threadIdx.x; i < N; i += gridDim.x * 256) {
        int n = i / SP4; int r = i - n * SP4;
        int pw = r / SP3; r -= pw * SP3;
        int px = r / SP2; r -= px * SP2;
        int py = r / SP;  int pz = r - py * SP;
        float v = 0.f;
        if (pw >= 1 && pw <= S && px >= 1 && px <= S &&
            py >= 1 && py <= S && pz >= 1 && pz <= S)
            v = x[n * S4 + (pw - 1) * S3 + (px - 1) * S2 + (py - 1) * S + (pz - 1)];
        xp[i] = v;
    }
}

// ---------------------------------------------------------------------------
// Layer 1: Cin=1 -> Cout=16, K dimension = taps (81 padded to 84 with zero
// weights).  Reads padded x: no masks anywhere.  Output written channel-pair
// interleaved into padded actA: act[(co>>1)*SP4 + pos]*2 + (co&1).
// ---------------------------------------------------------------------------
__global__ __launch_bounds__(128)
void conv4d_l1(const float* __restrict__ xp, const float* __restrict__ w1,
               const float* __restrict__ b1, float* __restrict__ act, int perm)
{
    __shared__ float wl[84 * 16];   // pair-interleaved: ((t>>1)*16+co)*2 + (t&1)
    __shared__ int2  tof[42];       // padded-grid offsets for tap pair (2p, 2p+1)
    __shared__ float bl[16];

    const int tid = threadIdx.x;
    for (int i = tid; i < 84 * 16; i += 128) {
        int t = i >> 4, co = i & 15;
        float v = 0.f;
        if (t < 81) v = w1[co * 81 + perm_tap(t, perm)];
        wl[(((t >> 1) * 16 + co) << 1) + (t & 1)] = v;
    }
    for (int p = tid; p < 42; p += 128) {
        int o[2];
        for (int h = 0; h < 2; ++h) {
            int t = 2 * p + h;
            if (t < 81) {
                int dz = t % 3, dy = (t / 3) % 3, dx = (t / 9) % 3, dw = t / 27;
                o[h] = dw * SP3 + dx * SP2 + dy * SP + dz;
            } else {
                o[h] = 0;   // zero A-weight makes this contribution 0 anyway
            }
        }
        tof[p] = make_int2(o[0], o[1]);
    }
    if (tid < 16) bl[tid] = b1[tid];
    __syncthreads();

    const int wave = tid >> 5, lane = tid & 31;
    const int co = lane & 15;                 // A row (M) and B column (N)
    const int khalf = (lane >> 4) << 1;       // lanes 16-31 carry K=2,3
    const int W = blockIdx.x / S, X = blockIdx.x % S;
    const int sbase = W * SP3 + X * SP2;      // padded input corner for dw=dx=0

    for (int k = 0; k < 9; ++k) {
        const int T = wave + 4 * k;           // constant trip count: no exec loop
        const int q0 = T * 16;
        const int q = q0 + co;
        const int y = q / S, z = q % S;
        const int lyz = y * SP + z;
        const int base = sbase + lyz;
        v8f acc = {};
        #pragma unroll 3
        for (int G = 0; G < 21; ++G) {
            const int t0 = 4 * G + khalf;     // even
            v2f a = *(const v2f*)(wl + ((((t0 >> 1) * 16) + co) << 1));
            int2 tf = tof[t0 >> 1];
            v2f b;
            b.x = xp[base + tf.x];
            b.y = xp[base + tf.y];
            acc = __builtin_amdgcn_wmma_f32_16x16x4_f32(
                      false, a, false, b, (short)0, acc, false, false);
        }
        const int mbase = (lane >> 4) << 3;
        const int opos = base + (SP3 + SP2 + SP + 1);   // padded interior position
        #pragma unroll
        for (int r = 0; r < 8; ++r) {
            int com = r + mbase;
            float v = acc[r] + bl[com];
            v = v > 0.f ? v : 0.f;
            act[(((com >> 1) * SP4 + opos) << 1) + (com & 1)] = v;
        }
    }
}

// ---------------------------------------------------------------------------
// Layer 2: Cin=16 -> Cout=16 (the hot layer).  Padded, pair-interleaved input:
// per tap, 4 WMMA f32_16x16x4 per tile; each wave accumulates 3 tiles so each
// A ds_load_b64 feeds 3 WMMAs; each B operand is a single global_load_b64.
// ---------------------------------------------------------------------------
__global__ __launch_bounds__(128)
void conv4d_l2(const float* __restrict__ in, const float* __restrict__ w2,
               const float* __restrict__ b2, float* __restrict__ out, int perm)
{
    extern __shared__ float wl2[];  // 81*256: ((t*8 + ci/2)*16 + co)*2 + (ci&1)
    __shared__ float bl[16];

    const int tid = threadIdx.x;
    for (int i = tid; i < 81 * 256; i += 128) {
        int t = i >> 8, rem = i & 255, ci = rem >> 4, co = rem & 15;
        wl2[((((t * 8) + (ci >> 1)) * 16 + co) << 1) + (ci & 1)] =
            w2[(co * 16 + ci) * 81 + perm_tap(t, perm)];
    }
    if (tid < 16) bl[tid] = b2[tid];
    __syncthreads();

    const int wave = tid >> 5, lane = tid & 31;
    const int co = lane & 15;
    const int hihalf = lane >> 4;             // upper lanes carry K = ci0+2,ci0+3
    const int koff = hihalf * (2 * SP4);      // pair-offset for upper half
    const int W = blockIdx.x / S, X = blockIdx.x % S;

    for (int gi = 0; gi < 3; ++gi) {
        const int q0 = (wave * 9 + gi * 3) * 16;
        int lb0, lb1, lb2;
        {
            int q = q0 + co;        lb0 = (q / S) * SP + (q % S);
            q = q0 + 16 + co;       lb1 = (q / S) * SP + (q % S);
            q = q0 + 32 + co;       lb2 = (q / S) * SP + (q % S);
        }
        v8f acc0 = {}, acc1 = {}, acc2 = {};
        for (int dwx = 0; dwx < 9; ++dwx) {
            const int dw = dwx / 3, dx = dwx - dw * 3;
            const int splane = (W + dw) * SP3 + (X + dx) * SP2;   // scalar
            for (int dyz = 0; dyz < 9; ++dyz) {
                const int dy = dyz / 3, dz = dyz - dy * 3;
                const int so = (splane + dy * SP + dz) << 1;      // scalar, paired
                const int t = dwx * 9 + dyz;
                const float* wt = wl2 + t * 256;
                const int o0 = so + (lb0 << 1) + koff;
                const int o1 = so + (lb1 << 1) + koff;
                const int o2 = so + (lb2 << 1) + koff;
                #pragma unroll
                for (int g = 0; g < 4; ++g) {
                    const int cp0 = 2 * g + hihalf;     // channel-pair index
                    v2f a = *(const v2f*)(wt + (((cp0 * 16) + co) << 1));
                    const int go = g * (4 * SP4);
                    v2f b0 = *(const v2f*)(in + o0 + go);
                    v2f b1 = *(const v2f*)(in + o1 + go);
                    v2f b2v = *(const v2f*)(in + o2 + go);
                    acc0 = __builtin_amdgcn_wmma_f32_16x16x4_f32(
                               false, a, false, b0, (short)0, acc0, false, false);
                    acc1 = __builtin_amdgcn_wmma_f32_16x16x4_f32(
                               false, a, false, b1, (short)0, acc1, false, false);
                    acc2 = __builtin_amdgcn_wmma_f32_16x16x4_f32(
                               false, a, false, b2v, (short)0, acc2, false, false);
                }
            }
        }
        const int mbase = hihalf << 3;
        const int obase = W * SP3 + X * SP2 + (SP3 + SP2 + SP + 1);
        #pragma unroll
        for (int r = 0; r < 8; ++r) {
            const int com = r + mbase;
            const int cb = ((com >> 1) * SP4) << 1;
            const int h = com & 1;
            float v0 = acc0[r] + bl[com];
            float v1 = acc1[r] + bl[com];
            float v2s = acc2[r] + bl[com];
            out[cb + ((obase + lb0) << 1) + h] = v0 > 0.f ? v0 : 0.f;
            out[cb + ((obase + lb1) << 1) + h] = v1 > 0.f ? v1 : 0.f;
            out[cb + ((obase + lb2) << 1) + h] = v2s > 0.f ? v2s : 0.f;
        }
    }
}

// ---------------------------------------------------------------------------
// Layer 3: Cin=16 -> Cout=1 (~5% of FLOPs): VALU FMAs, padded pair-interleaved
// input -> no masks, b64 loads.  br0 stores, br1 accumulates into d_out.
// ---------------------------------------------------------------------------
__global__ __launch_bounds__(256)
void conv4d_l3(const float* __restrict__ in, const float* __restrict__ w3,
               const float* __restrict__ b3, float* __restrict__ out,
               int perm, int accumulate)
{
    __shared__ float wl[16 * 81];
    const int tid = threadIdx.x;
    for (int i = tid; i < 16 * 81; i += 256) {
        int ci = i / 81, t = i - ci * 81;
        wl[ci * 81 + t] = w3[ci * 81 + perm_tap(t, perm)];
    }
    __syncthreads();

    const int pos = blockIdx.x * 256 + tid;
    const int z = pos % S, y = (pos / S) % S, X = (pos / S2) % S, W = pos / S3;
    const int base0 = W * SP3 + X * SP2 + y * SP + z;

    float sum = b3[0];
    for (int dwx = 0; dwx < 9; ++dwx) {
        const int dw = dwx / 3, dx = dwx - dw * 3;
        const int sp = dw * SP3 + dx * SP2;
        for (int dyz = 0; dyz < 9; ++dyz) {
            const int dy = dyz / 3, dz = dyz - dy * 3;
            const int t = dwx * 9 + dyz;
            const int o2 = (base0 + sp + dy * SP + dz) << 1;
            #pragma unroll
            for (int cp = 0; cp < 8; ++cp) {
                v2f v = *(const v2f*)(in + o2 + cp * (2 * SP4));
                sum = fmaf(v.x, wl[(2 * cp) * 81 + t],
                      fmaf(v.y, wl[(2 * cp + 1) * 81 + t], sum));
            }
        }
    }
    float r = sum > 0.f ? sum : 0.f;
    if (accumulate) out[pos] += r;
    else            out[pos]  = r;
}

// ---------------------------------------------------------------------------
extern "C" void kernel_launch(void* const* d_in, const int* in_sizes, int n_in,
                              void* d_out, int out_size, void* d_ws, size_t ws_size,
                              hipStream_t stream)
{
    (void)in_sizes; (void)n_in; (void)out_size; (void)ws_size;

    const float* x  = (const float*)d_in[0];
    const float* w1 = (const float*)d_in[1];
    const float* b1 = (const float*)d_in[2];
    const float* w2 = (const float*)d_in[3];
    const float* b2 = (const float*)d_in[4];
    const float* w3 = (const float*)d_in[5];
    const float* b3 = (const float*)d_in[6];
    float* out = (float*)d_out;

    // Workspace: xpad (8*SP4) | actA (16*SP4, paired) | actB (16*SP4, paired)
    float* xpad = (float*)d_ws;
    float* actA = xpad + (size_t)8 * SP4;
    float* actB = actA + (size_t)16 * SP4;

    // Initialize padded buffers (halos must be zero; ws is poisoned by harness).
    zero_ws<<<2048, 256, 0, stream>>>((float4*)actA, (32 * SP4) / 4);
    pad_x<<<2048, 256, 0, stream>>>(x, xpad);

    // out = net_w(x) + net_{w'}(x)  with w'[..,dw,dx,dy,dz] = w[..,dy,dz,dw,dx]
    for (int br = 0; br < 2; ++br) {
        for (int n = 0; n < 8; ++n) {
            conv4d_l1<<<S * S, 128, 0, stream>>>(xpad + (size_t)n * SP4, w1, b1, actA, br);
            conv4d_l2<<<S * S, 128, 81 * 256 * sizeof(float), stream>>>(actA, w2, b2, actB, br);
            conv4d_l3<<<S4 / 256, 256, 0, stream>>>(actB, w3, b3, out + (size_t)n * S4, br, br);
        }
    }
}